// SparseGConv2d_54580444397642
// MI455X (gfx1250) — compile-verified
//
#include <hip/hip_runtime.h>
#include <hip/hip_bf16.h>

// Problem constants (match reference)
#define NN 16384      // nodes
#define NB 256        // batch
#define NE 262144     // edges
#define KC 4          // chebyshev order
#define FO 32         // output features

typedef __attribute__((ext_vector_type(2))) float v2f;
typedef __attribute__((ext_vector_type(8))) float v8f;

// ---------------------------------------------------------------------------
// Kernel 1: transpose x (B, N) -> t0 (N, B), LDS-tiled for coalescing.
// ---------------------------------------------------------------------------
__global__ void k_transpose(const float* __restrict__ x, float* __restrict__ t0) {
    __shared__ float tile[32][33];
    const int bx = blockIdx.x * 32;          // node-dim base
    const int by = blockIdx.y * 32;          // batch-dim base
    const int tx = threadIdx.x;              // 0..31
    const int ty = threadIdx.y;              // 0..7
#pragma unroll
    for (int i = 0; i < 32; i += 8) {
        tile[ty + i][tx] = x[(size_t)(by + ty + i) * NN + (bx + tx)];
    }
    __syncthreads();
#pragma unroll
    for (int i = 0; i < 32; i += 8) {
        t0[(size_t)(bx + ty + i) * NB + (by + tx)] = tile[tx][ty + i];
    }
}

// ---------------------------------------------------------------------------
// Kernel 2: elementwise  o = ca*a + cb*b   (float4 vectorized)
// Used to seed t_k = -2*t_{k-1} - t_{k-2} before the atomic SpMM pass.
// ---------------------------------------------------------------------------
__global__ void k_combine(const float4* __restrict__ a, const float4* __restrict__ b,
                          float4* __restrict__ o, float ca, float cb) {
    size_t i = (size_t)blockIdx.x * blockDim.x + threadIdx.x;
    float4 va = a[i];
    float4 vb = b[i];
    float4 r;
    r.x = ca * va.x + cb * vb.x;
    r.y = ca * va.y + cb * vb.y;
    r.z = ca * va.z + cb * vb.z;
    r.w = ca * va.w + cb * vb.w;
    o[i] = r;
}

// ---------------------------------------------------------------------------
// Kernel 3: SpMM scatter  t_out[row, :] += s * vals[e] * t_in[col, :]
// One block per edge, one lane per batch column: coalesced gather + coalesced
// fp32 global atomics (L2-resident: each term matrix is only 16 MB).
// ---------------------------------------------------------------------------
__global__ void k_spmm(const float* __restrict__ t_in, float* __restrict__ t_out,
                       const float* __restrict__ vals, const int* __restrict__ rows,
                       const int* __restrict__ cols, float s) {
    const int e = blockIdx.x;
    const int c = threadIdx.x;            // 0..255 batch column
    const int r = rows[e];                // scalar-cached broadcast
    const int g = cols[e];
    const float v = s * vals[e];
    const float contrib = v * t_in[(size_t)g * NB + c];
    unsafeAtomicAdd(&t_out[(size_t)r * NB + c], contrib);   // global_atomic_add_f32
}

// ---------------------------------------------------------------------------
// Kernel 4: final linear via V_WMMA_F32_16X16X4_F32.
//   feats[i, k] = t_k[n, b]  with i = b*N + n  (16 consecutive i per tile,
//   all sharing b since N % 16 == 0)
//   out[i, f] = sum_k feats[i,k] * weight[f,k] + bias[f],  f in [0,32)
// A (16x4 f32, 2 VGPRs): lanes 0-15 -> {K0,K1}, lanes 16-31 -> {K2,K3} (ISA layout)
// B (4x16 f32, 2 VGPRs): mirrored K striping, N = lane & 15
// C/D (16x16 f32, 8 VGPRs): VGPR r -> row M=r (lanes 0-15) / M=r+8 (lanes 16-31)
// One tile (16 rows x 32 outs) per wave, 2 WMMA per wave, exact grid cover so
// EXEC is all ones at the WMMA (hardware requirement).
// ---------------------------------------------------------------------------
__global__ void k_linear_wmma(const float* __restrict__ t0, const float* __restrict__ t1,
                              const float* __restrict__ t2, const float* __restrict__ t3,
                              const float* __restrict__ weight, const float* __restrict__ bias,
                              float* __restrict__ out) {
    const int lane = threadIdx.x & 31;
    const int wave = threadIdx.x >> 5;
    const int tile = blockIdx.x * 8 + wave;           // [0, B*N/16)
    const size_t i0 = (size_t)tile * 16;              // base flattened row (b*N + n)
    const int b = (int)(i0 / NN);
    const int n0 = (int)(i0 % NN);

    // ---- A operand: 16 rows x 4 chebyshev features ----
    const int m = lane & 15;
    const bool hi = lane >= 16;
    const float* pa0 = hi ? t2 : t0;                  // K = hi ? 2 : 0
    const float* pa1 = hi ? t3 : t1;                  // K = hi ? 3 : 1
    const size_t aidx = (size_t)(n0 + m) * NB + b;
    v2f a;
    a.x = pa0[aidx];
    a.y = pa1[aidx];

    // ---- two output-feature tiles of 16 ----
    const int kbase = (lane >> 4) * 2;                // 0 or 2
#pragma unroll
    for (int tf = 0; tf < 2; ++tf) {
        const int f = (lane & 15) + 16 * tf;
        v2f bm;
        bm.x = weight[f * KC + kbase];                // B[k=kbase  ][n=f]
        bm.y = weight[f * KC + kbase + 1];            // B[k=kbase+1][n=f]

        const float bv = bias[f];
        v8f c;
#pragma unroll
        for (int j = 0; j < 8; ++j) c[j] = bv;

        v8f d = __builtin_amdgcn_wmma_f32_16x16x4_f32(
            /*neg_a=*/false, a, /*neg_b=*/false, bm,
            /*c_mod=*/(short)0, c, /*reuse_a=*/false, /*reuse_b=*/false);

        // ---- store: VGPR j holds row (j | j+8), col = (lane&15) + 16*tf ----
        const int col = (lane & 15) + 16 * tf;
        const int rbase = hi ? 8 : 0;
#pragma unroll
        for (int j = 0; j < 8; ++j) {
            out[(i0 + rbase + j) * FO + col] = d[j];
        }
    }
}

// ---------------------------------------------------------------------------
// Host launcher
// ---------------------------------------------------------------------------
extern "C" void kernel_launch(void* const* d_in, const int* in_sizes, int n_in,
                              void* d_out, int out_size, void* d_ws, size_t ws_size,
                              hipStream_t stream) {
    const float* x      = (const float*)d_in[0];   // (B, N)
    const float* vals   = (const float*)d_in[1];   // (E,)
    const int*   rows   = (const int*)  d_in[2];   // (E,)
    const int*   cols   = (const int*)  d_in[3];   // (E,)
    const float* weight = (const float*)d_in[4];   // (32, 4)
    const float* bias   = (const float*)d_in[5];   // (32,)
    float* out = (float*)d_out;                    // (B, N, 32)

    const size_t nb_elems = (size_t)NN * NB;       // 4,194,304 floats = 16 MB
    float* t0 = (float*)d_ws;
    float* t1 = t0 + nb_elems;
    float* t2 = t1 + nb_elems;
    float* t3 = t2 + nb_elems;                     // total 64 MB of workspace

    // 1) t0 = x^T
    dim3 tb(32, 8);
    dim3 tg(NN / 32, NB / 32);
    k_transpose<<<tg, tb, 0, stream>>>(x, t0);

    const int cblocks = (int)(nb_elems / 4 / 256); // float4-vectorized elementwise
    // 2) t1 = -t0 ; then t1 += spmm(t0)            => t1 = Lhat t0
    k_combine<<<cblocks, 256, 0, stream>>>((const float4*)t0, (const float4*)t0,
                                           (float4*)t1, -1.0f, 0.0f);
    k_spmm<<<NE, NB, 0, stream>>>(t0, t1, vals, rows, cols, 1.0f);

    // 3) t2 = -2*t1 - t0 ; t2 += 2*spmm(t1)        => t2 = 2 Lhat t1 - t0
    k_combine<<<cblocks, 256, 0, stream>>>((const float4*)t1, (const float4*)t0,
                                           (float4*)t2, -2.0f, -1.0f);
    k_spmm<<<NE, NB, 0, stream>>>(t1, t2, vals, rows, cols, 2.0f);

    // 4) t3 = -2*t2 - t1 ; t3 += 2*spmm(t2)        => t3 = 2 Lhat t2 - t1
    k_combine<<<cblocks, 256, 0, stream>>>((const float4*)t2, (const float4*)t1,
                                           (float4*)t3, -2.0f, -1.0f);
    k_spmm<<<NE, NB, 0, stream>>>(t2, t3, vals, rows, cols, 2.0f);

    // 5) out = feats @ W^T + bias via WMMA f32 16x16x4
    const int tiles = (int)(nb_elems / 16);        // 262144 row tiles
    k_linear_wmma<<<tiles / 8, 256, 0, stream>>>(t0, t1, t2, t3, weight, bias, out);
}